// MinimalSSMTorch_20976620274247
// MI455X (gfx1250) — compile-verified
//
#include <hip/hip_runtime.h>
#include <math.h>

// Problem constants (from reference)
#define BATCH   4
#define SEQ     2048
#define DMODEL  1024
#define DINNER  2048
#define NSTATE  16
#define MROWS   (BATCH*SEQ)      // 8192
#define WIN_N   (2*DINNER)       // 4096

typedef float v2f __attribute__((ext_vector_type(2)));
typedef float v8f __attribute__((ext_vector_type(8)));

static constexpr int BM   = 128;
static constexpr int BN   = 64;
static constexpr int BK   = 16;
static constexpr int LDAS = BK + 4;   // 20: float4-aligned, bank-spread
static constexpr int LDBS = BK + 2;   // 18: even (ds_load_b64-aligned)

__device__ __forceinline__ v8f wmma_f32(v2f a, v2f b, v8f c) {
  // V_WMMA_F32_16X16X4_F32 : D = A(16x4) * B(4x16) + C, full fp32
  return __builtin_amdgcn_wmma_f32_16x16x4_f32(false, a, false, b, (short)0, c,
                                               false, false);
}
__device__ __forceinline__ v2f frag2(const float* p) {
  v2f r; r[0] = p[0]; r[1] = p[1]; return r;
}

// ---------------------------------------------------------------------------
// Kernel 0: pack W_A|W_B|W_C [2048,16] each -> Wabc [2048,48]
// ---------------------------------------------------------------------------
__global__ __launch_bounds__(256) void pack_wabc(
    const float* __restrict__ WA, const float* __restrict__ WB,
    const float* __restrict__ WC, float* __restrict__ Wabc) {
  int i = blockIdx.x * 256 + threadIdx.x;
  if (i >= DINNER * NSTATE) return;
  int k = i >> 4, n = i & 15;
  Wabc[k * 48 +      n] = WA[i];
  Wabc[k * 48 + 16 + n] = WB[i];
  Wabc[k * 48 + 32 + n] = WC[i];
}

// ---------------------------------------------------------------------------
// Kernel 1: x[8192,1024] @ W_in[1024,4096], fused silu(u)*sigmoid(g)
//           -> Xin[8192,2048].  Block tile 128x64; each block computes both
//           column halves (n0 and n0+2048) so gating fuses in registers.
// ---------------------------------------------------------------------------
__global__ __launch_bounds__(256) void gemm1_gated(
    const float* __restrict__ X, const float* __restrict__ Win,
    float* __restrict__ Xin) {
  __shared__ __align__(16) float As[BM * LDAS];
  __shared__ __align__(16) float Bs[2][BN * LDBS];   // transposed [n][k]
  const int tid  = threadIdx.x;
  const int lane = tid & 31;
  const int wave = tid >> 5;          // 8 waves: 4 (m) x 2 (n)
  const int wm   = wave >> 1;
  const int wn   = wave & 1;
  const int r0   = blockIdx.x * BM;
  const int n0   = blockIdx.y * BN;

  const v8f vz = {0.f,0.f,0.f,0.f,0.f,0.f,0.f,0.f};
  v8f acc[2][2][2];
  for (int s = 0; s < 2; s++)
    for (int i = 0; i < 2; i++)
      for (int j = 0; j < 2; j++) acc[s][i][j] = vz;

  for (int kk = 0; kk < DMODEL; kk += BK) {
    // A tile: 128x16 (512 float4, 2 per thread)
#pragma unroll
    for (int i = 0; i < 2; i++) {
      int idx = tid + i * 256;
      int row = idx >> 2, c4 = idx & 3;
      float4 v = *(const float4*)(X + (size_t)(r0 + row) * DMODEL + kk + c4 * 4);
      *(float4*)(&As[row * LDAS + c4 * 4]) = v;
    }
    // B tiles (both halves), store transposed [n][k]
#pragma unroll
    for (int s = 0; s < 2; s++) {
      int krow = tid >> 4;            // 0..15
      int c4   = tid & 15;            // 0..15
      float4 v = *(const float4*)(Win + (size_t)(kk + krow) * WIN_N +
                                  n0 + s * DINNER + c4 * 4);
      float* bp = &Bs[s][(c4 * 4) * LDBS + krow];
      bp[0 * LDBS] = v.x; bp[1 * LDBS] = v.y;
      bp[2 * LDBS] = v.z; bp[3 * LDBS] = v.w;
    }
    __syncthreads();
#pragma unroll
    for (int ks = 0; ks < 4; ks++) {
      const int kb = ks * 4 + ((lane >> 4) << 1);
      const int ln = lane & 15;
      v2f a[2], b[2][2];
#pragma unroll
      for (int i = 0; i < 2; i++)
        a[i] = frag2(&As[(wm * 32 + i * 16 + ln) * LDAS + kb]);
#pragma unroll
      for (int s = 0; s < 2; s++)
#pragma unroll
        for (int j = 0; j < 2; j++)
          b[s][j] = frag2(&Bs[s][(wn * 32 + j * 16 + ln) * LDBS + kb]);
#pragma unroll
      for (int s = 0; s < 2; s++)
#pragma unroll
        for (int i = 0; i < 2; i++)
#pragma unroll
          for (int j = 0; j < 2; j++)
            acc[s][i][j] = wmma_f32(a[i], b[s][j], acc[s][i][j]);
    }
    __syncthreads();
  }
  // Epilogue: x_in = silu(u) * sigmoid(g)
  const int ncol = lane & 15;
  const int mhi  = (lane >> 4) * 8;
#pragma unroll
  for (int i = 0; i < 2; i++)
#pragma unroll
    for (int j = 0; j < 2; j++)
#pragma unroll
      for (int vv = 0; vv < 8; vv++) {
        float u  = acc[0][i][j][vv];
        float g  = acc[1][i][j][vv];
        float su = u / (1.f + expf(-u));
        float sg = 1.f / (1.f + expf(-g));
        int row = r0 + wm * 32 + i * 16 + mhi + vv;
        int col = n0 + wn * 32 + j * 16 + ncol;
        Xin[(size_t)row * DINNER + col] = su * sg;
      }
}

// ---------------------------------------------------------------------------
// Kernel 2: Xin[8192,2048] @ Wabc[2048,48] -> ABC[8192,48]
//           cols 0..15 transformed to decay = exp(-exp(clip(v,-5,0)))
// ---------------------------------------------------------------------------
__global__ __launch_bounds__(256) void gemm2_abc(
    const float* __restrict__ Xin, const float* __restrict__ Wabc,
    float* __restrict__ ABC) {
  __shared__ __align__(16) float As[BM * LDAS];
  __shared__ __align__(16) float Bs[48 * LDBS];     // transposed [n][k]
  const int tid  = threadIdx.x;
  const int lane = tid & 31;
  const int wave = tid >> 5;          // each wave: 16 rows x 48 cols
  const int r0   = blockIdx.x * BM;

  const v8f vz = {0.f,0.f,0.f,0.f,0.f,0.f,0.f,0.f};
  v8f acc[3] = {vz, vz, vz};

  for (int kk = 0; kk < DINNER; kk += BK) {
#pragma unroll
    for (int i = 0; i < 2; i++) {
      int idx = tid + i * 256;
      int row = idx >> 2, c4 = idx & 3;
      float4 v = *(const float4*)(Xin + (size_t)(r0 + row) * DINNER + kk + c4 * 4);
      *(float4*)(&As[row * LDAS + c4 * 4]) = v;
    }
#pragma unroll
    for (int i = 0; i < 3; i++) {
      int idx  = tid + i * 256;       // 0..767
      int krow = idx / 48;
      int c    = idx - krow * 48;
      Bs[c * LDBS + krow] = Wabc[(size_t)(kk + krow) * 48 + c];
    }
    __syncthreads();
#pragma unroll
    for (int ks = 0; ks < 4; ks++) {
      const int kb = ks * 4 + ((lane >> 4) << 1);
      const int ln = lane & 15;
      v2f a = frag2(&As[(wave * 16 + ln) * LDAS + kb]);
#pragma unroll
      for (int j = 0; j < 3; j++) {
        v2f b = frag2(&Bs[(j * 16 + ln) * LDBS + kb]);
        acc[j] = wmma_f32(a, b, acc[j]);
      }
    }
    __syncthreads();
  }
  const int ncol = lane & 15;
  const int mhi  = (lane >> 4) * 8;
#pragma unroll
  for (int j = 0; j < 3; j++)
#pragma unroll
    for (int vv = 0; vv < 8; vv++) {
      float val = acc[j][vv];
      if (j == 0) {                   // decay = exp(A), A = -exp(clip(v,-5,0))
        float c = fminf(fmaxf(val, -5.f), 0.f);
        val = expf(-expf(c));
      }
      int row = r0 + wave * 16 + mhi + vv;
      ABC[(size_t)row * 48 + j * 16 + ncol] = val;
    }
}

// ---------------------------------------------------------------------------
// Kernel 3: sequential SSM scan over T.  One thread per (b,d); 16 states in
//           registers; decay/B/C staged in LDS per 32-step chunk.  The 32
//           x-values of a chunk are batch-loaded into registers BEFORE any
//           store, so the in-place read/write on XO cannot serialize the
//           memory pipeline on (unprovable) aliasing; one latency per chunk.
// ---------------------------------------------------------------------------
#define TCHUNK 32
__global__ __launch_bounds__(256) void ssm_scan(
    const float* __restrict__ ABC, float* __restrict__ XO) {
  __shared__ __align__(16) float s_abc[TCHUNK * 48];
  const int tid = threadIdx.x;
  const int d   = blockIdx.x * 256 + tid;
  const size_t baseRow = (size_t)blockIdx.y * SEQ;

  float st[NSTATE];
#pragma unroll
  for (int n = 0; n < NSTATE; n++) st[n] = 0.f;

  for (int t0 = 0; t0 < SEQ; t0 += TCHUNK) {
    __syncthreads();                  // prior chunk's s_abc reads done
#pragma unroll
    for (int i = 0; i < 6; i++) {
      int idx = tid + i * 256;        // 0..1535
      s_abc[idx] = ABC[(baseRow + t0) * 48 + idx];
    }
    // Batch-load this chunk's x values (overlaps with LDS fill + barrier).
    float* xp = XO + (baseRow + t0) * DINNER + d;
    float xv[TCHUNK];
#pragma unroll
    for (int tt = 0; tt < TCHUNK; tt++)
      xv[tt] = xp[(size_t)tt * DINNER];
    __syncthreads();
#pragma unroll
    for (int tt = 0; tt < TCHUNK; tt++) {
      const float4* rp = (const float4*)&s_abc[tt * 48];
      float x = xv[tt];
      float acc = 0.f;
#pragma unroll
      for (int q = 0; q < 4; q++) {
        float4 de = rp[q], bb = rp[4 + q], cc = rp[8 + q];
        int n = q * 4;
        st[n+0] = st[n+0] * de.x + x * bb.x;  acc += st[n+0] * cc.x;
        st[n+1] = st[n+1] * de.y + x * bb.y;  acc += st[n+1] * cc.y;
        st[n+2] = st[n+2] * de.z + x * bb.z;  acc += st[n+2] * cc.z;
        st[n+3] = st[n+3] * de.w + x * bb.w;  acc += st[n+3] * cc.w;
      }
      xv[tt] = acc;                   // recycle register
    }
    // Batch-store the chunk's outputs (in place over Xin).
#pragma unroll
    for (int tt = 0; tt < TCHUNK; tt++)
      xp[(size_t)tt * DINNER] = xv[tt];
  }
}

// ---------------------------------------------------------------------------
// Kernel 4: per-row RMS scale = rsqrt(mean(x^2) + eps)
// ---------------------------------------------------------------------------
__global__ __launch_bounds__(256) void rms_scale(
    const float* __restrict__ XO, float* __restrict__ scale) {
  __shared__ float red[256];
  const int row = blockIdx.x;
  const int tid = threadIdx.x;
  const float* p = XO + (size_t)row * DINNER;
  float s = 0.f;
#pragma unroll
  for (int i = 0; i < 8; i++) {
    float v = p[tid + i * 256];
    s += v * v;
  }
  red[tid] = s;
  __syncthreads();
  for (int off = 128; off > 0; off >>= 1) {
    if (tid < off) red[tid] += red[tid + off];
    __syncthreads();
  }
  if (tid == 0)
    scale[row] = rsqrtf(red[0] * (1.f / DINNER) + 1.1920929e-7f);
}

// ---------------------------------------------------------------------------
// Kernel 5: Out = (XO * rowscale * norm_w) @ W_out[2048,1024]
//           RMSNorm fused into the A-tile LDS fill.
// ---------------------------------------------------------------------------
__global__ __launch_bounds__(256) void gemm3_out(
    const float* __restrict__ XO, const float* __restrict__ rowscale,
    const float* __restrict__ normw, const float* __restrict__ Wout,
    float* __restrict__ Out) {
  __shared__ __align__(16) float As[BM * LDAS];
  __shared__ __align__(16) float Bs[BN * LDBS];
  const int tid  = threadIdx.x;
  const int lane = tid & 31;
  const int wave = tid >> 5;
  const int wm   = wave >> 1;
  const int wn   = wave & 1;
  const int r0   = blockIdx.x * BM;
  const int n0   = blockIdx.y * BN;

  const v8f vz = {0.f,0.f,0.f,0.f,0.f,0.f,0.f,0.f};
  v8f acc[2][2];
  for (int i = 0; i < 2; i++)
    for (int j = 0; j < 2; j++) acc[i][j] = vz;

  for (int kk = 0; kk < DINNER; kk += BK) {
#pragma unroll
    for (int i = 0; i < 2; i++) {
      int idx = tid + i * 256;
      int row = idx >> 2, c4 = idx & 3;
      float4 v  = *(const float4*)(XO + (size_t)(r0 + row) * DINNER + kk + c4 * 4);
      float4 w  = *(const float4*)(normw + kk + c4 * 4);
      float  rs = rowscale[r0 + row];
      v.x *= rs * w.x; v.y *= rs * w.y; v.z *= rs * w.z; v.w *= rs * w.w;
      *(float4*)(&As[row * LDAS + c4 * 4]) = v;
    }
    {
      int krow = tid >> 4, c4 = tid & 15;
      float4 v = *(const float4*)(Wout + (size_t)(kk + krow) * DMODEL + n0 + c4 * 4);
      float* bp = &Bs[(c4 * 4) * LDBS + krow];
      bp[0 * LDBS] = v.x; bp[1 * LDBS] = v.y;
      bp[2 * LDBS] = v.z; bp[3 * LDBS] = v.w;
    }
    __syncthreads();
#pragma unroll
    for (int ks = 0; ks < 4; ks++) {
      const int kb = ks * 4 + ((lane >> 4) << 1);
      const int ln = lane & 15;
      v2f a[2], b[2];
#pragma unroll
      for (int i = 0; i < 2; i++)
        a[i] = frag2(&As[(wm * 32 + i * 16 + ln) * LDAS + kb]);
#pragma unroll
      for (int j = 0; j < 2; j++)
        b[j] = frag2(&Bs[(wn * 32 + j * 16 + ln) * LDBS + kb]);
#pragma unroll
      for (int i = 0; i < 2; i++)
#pragma unroll
        for (int j = 0; j < 2; j++)
          acc[i][j] = wmma_f32(a[i], b[j], acc[i][j]);
    }
    __syncthreads();
  }
  const int ncol = lane & 15;
  const int mhi  = (lane >> 4) * 8;
#pragma unroll
  for (int i = 0; i < 2; i++)
#pragma unroll
    for (int j = 0; j < 2; j++)
#pragma unroll
      for (int vv = 0; vv < 8; vv++) {
        int row = r0 + wm * 32 + i * 16 + mhi + vv;
        int col = n0 + wn * 32 + j * 16 + ncol;
        Out[(size_t)row * DMODEL + col] = acc[i][j][vv];
      }
}

// ---------------------------------------------------------------------------
extern "C" void kernel_launch(void* const* d_in, const int* in_sizes, int n_in,
                              void* d_out, int out_size, void* d_ws,
                              size_t ws_size, hipStream_t stream) {
  const float* x     = (const float*)d_in[0];
  const float* Win   = (const float*)d_in[1];
  const float* WA    = (const float*)d_in[2];
  const float* WB    = (const float*)d_in[3];
  const float* WC    = (const float*)d_in[4];
  const float* Wout  = (const float*)d_in[5];
  const float* normw = (const float*)d_in[6];
  float* out = (float*)d_out;

  char* ws = (char*)d_ws;
  size_t off = 0;
  auto take = [&](size_t bytes) {
    void* p = ws + off;
    off = (off + bytes + 255) & ~(size_t)255;
    return p;
  };
  float* xin      = (float*)take((size_t)MROWS * DINNER * 4);   // 64 MiB, reused as scan output
  float* abc      = (float*)take((size_t)MROWS * 48 * 4);
  float* wabc     = (float*)take((size_t)DINNER * 48 * 4);
  float* rowscale = (float*)take((size_t)MROWS * 4);
  (void)ws_size; (void)in_sizes; (void)n_in; (void)out_size;

  pack_wabc<<<(DINNER * NSTATE + 255) / 256, 256, 0, stream>>>(WA, WB, WC, wabc);
  gemm1_gated<<<dim3(MROWS / BM, DINNER / BN), 256, 0, stream>>>(x, Win, xin);
  gemm2_abc<<<dim3(MROWS / BM), 256, 0, stream>>>(xin, wabc, abc);
  ssm_scan<<<dim3(DINNER / 256, BATCH), 256, 0, stream>>>(abc, xin);
  rms_scale<<<dim3(MROWS), 256, 0, stream>>>(xin, rowscale);
  gemm3_out<<<dim3(MROWS / BM, DMODEL / BN), 256, 0, stream>>>(
      xin, rowscale, normw, Wout, out);
}